// PMXELoss_27831388078897
// MI455X (gfx1250) — compile-verified
//
#include <hip/hip_runtime.h>
#include <math.h>

typedef __attribute__((ext_vector_type(2))) float v2f;
typedef __attribute__((ext_vector_type(8))) float v8f;

#define NQ   4096
#define NS   16384
#define DIM  128
#define NC   64
#define INFV 1000.0f

// ---------------------------------------------------------------------------
// Kernel A: per-class statistics. One block per class (64 blocks, 128 threads).
// mus[c][d] = sum of xs rows with label c ; SS[c] = sum of ||xs_j||^2 over class;
// cnts[c] = member count. No atomics -> deterministic.
// ---------------------------------------------------------------------------
__global__ void __launch_bounds__(128) class_stats_kernel(
    const float* __restrict__ xs, const int* __restrict__ ys,
    float* __restrict__ mus, float* __restrict__ SS, float* __restrict__ cnts) {
  __shared__ int   sy[2048];
  __shared__ float red[128];
  const int c = blockIdx.x;        // class id
  const int t = threadIdx.x;       // dimension owner, 0..127

  float acc = 0.0f, ssacc = 0.0f;
  int cnt = 0;
  for (int base = 0; base < NS; base += 2048) {
    for (int j = t; j < 2048; j += 128) sy[j] = ys[base + j];
    __syncthreads();
    for (int j = 0; j < 2048; ++j) {
      if (sy[j] == c) {                       // uniform branch per block
        float v = xs[(size_t)(base + j) * DIM + t];
        acc += v;
        ssacc += v * v;
        ++cnt;
      }
    }
    __syncthreads();
  }
  mus[(size_t)c * DIM + t] = acc;

  red[t] = ssacc;
  __syncthreads();
  for (int s = 64; s > 0; s >>= 1) {
    if (t < s) red[t] += red[t + s];
    __syncthreads();
  }
  if (t == 0) {
    SS[c]   = red[0];
    cnts[c] = (float)cnt;
  }
}

// ---------------------------------------------------------------------------
// Kernel B: main fused kernel. 256 blocks x 128 threads (4 wave32).
// Each block handles 16 queries; wave w computes the 16x16 WMMA tile
// G[q, c] = xq . mus^T for classes [16w, 16w+16), K = 128 via 32x
// v_wmma_f32_16x16x4_f32. Epilogue: analytic class sums, diag/-INF fix,
// cross-wave logsumexp, pos_logit, per-block partial sum (no atomics).
// ---------------------------------------------------------------------------
__global__ void __launch_bounds__(128) pmxe_main_kernel(
    const float* __restrict__ xq, const int* __restrict__ yq,
    const float* __restrict__ xs, const int* __restrict__ pos,
    const float* __restrict__ mus, const float* __restrict__ SS,
    const float* __restrict__ cnts, float* __restrict__ partial) {
  __shared__ float s_norm[16][NC + 1];     // +1 pad: conflict-free row scans
  __shared__ float s_part[16][8][4];
  __shared__ float s_qq[16], s_corr[16], s_plog[16], s_red[16];

  const int t     = threadIdx.x;
  const int qbase = blockIdx.x * 16;

  // ---- Phase 1: per-query scalars (all 128 threads: 16 queries x 8 chunks) ----
  {
    const int qi = t >> 3;                 // query within tile
    const int ch = t & 7;                  // 16-dim chunk
    const int q  = qbase + qi;
    const int p  = pos[q];
    const int yc = yq[q];
    const float cntc   = cnts[yc];
    const float invden = 1.0f / fmaxf(cntc - 1.0f, 0.1f);
    const float* xqr = xq  + (size_t)q  * DIM + ch * 16;
    const float* xsr = xs  + (size_t)p  * DIM + ch * 16;
    const float* mur = mus + (size_t)yc * DIM + ch * 16;
    float qq = 0.0f, dot = 0.0f, ssp = 0.0f, pd = 0.0f;
#pragma unroll
    for (int d = 0; d < 16; ++d) {
      float xv = xqr[d], sv = xsr[d], mv = mur[d];
      qq  += xv * xv;
      dot += xv * sv;
      ssp += sv * sv;
      float proto = (mv - xv) * invden;    // leave-one-out prototype
      float diff  = xv - proto;
      pd += diff * diff;
    }
    s_part[qi][ch][0] = qq;
    s_part[qi][ch][1] = dot;
    s_part[qi][ch][2] = ssp;
    s_part[qi][ch][3] = pd;
  }
  __syncthreads();
  if (t < 16) {
    float qq = 0.0f, dot = 0.0f, ssp = 0.0f, pd = 0.0f;
    for (int ch = 0; ch < 8; ++ch) {
      qq  += s_part[t][ch][0];
      dot += s_part[t][ch][1];
      ssp += s_part[t][ch][2];
      pd  += s_part[t][ch][3];
    }
    s_qq[t] = qq;
    const int q  = qbase + t;
    const int yc = yq[q];
    const float cntc = cnts[yc];
    // diagonal logit the reference zeroes out, then -INF where counts>1
    float diag = -0.5f * fmaxf(qq + ssp - 2.0f * dot, 0.0f);
    s_corr[t]  = -diag - ((cntc > 1.0f) ? INFV : 0.0f);
    s_plog[t]  = -sqrtf(fmaxf(pd, 0.0f));  // pos_logit
  }
  __syncthreads();

  // ---- Phase 2: WMMA f32 GEMM  G[16 x 64] = xq_tile[16x128] * mus^T[128x64] ----
  const int lane  = t & 31;
  const int w     = t >> 5;                // wave id 0..3
  const int cbase = w * 16;
  const int ln    = lane & 15;             // M (A) / N (B)
  const int khalf = (lane >> 4) * 2;       // lanes 0-15: K {k,k+1}; 16-31: {k+2,k+3}
  const float* arow = xq  + (size_t)(qbase + ln) * DIM + khalf;
  const float* brow = mus + (size_t)(cbase + ln) * DIM + khalf;

  v8f acc = {0.0f, 0.0f, 0.0f, 0.0f, 0.0f, 0.0f, 0.0f, 0.0f};
#pragma unroll
  for (int kk = 0; kk < 32; ++kk) {
    v2f a = *(const v2f*)(arow + kk * 4);
    v2f b = *(const v2f*)(brow + kk * 4);
    acc = __builtin_amdgcn_wmma_f32_16x16x4_f32(
        /*neg_a=*/false, a, /*neg_b=*/false, b,
        /*c_mod=*/(short)0, acc, /*reuse_a=*/false, /*reuse_b=*/false);
  }

  // ---- Epilogue: normalized[i,c] = -0.5(cnt*qq + SS - 2G)/ (cnt-1), + fixes ----
  {
    const int hi  = lane >> 4;
    const int cls = cbase + ln;
    const float cntc = cnts[cls];
    const float SSc  = SS[cls];
    const float inv  = 1.0f / (cntc - 1.0f);
#pragma unroll
    for (int r = 0; r < 8; ++r) {
      const int m = r + 8 * hi;            // C/D layout: vgpr r -> M=r / M=8+r
      const int q = qbase + m;
      float summed = -0.5f * (cntc * s_qq[m] + SSc - 2.0f * acc[r]);
      summed += (cls == yq[q]) ? s_corr[m] : 0.0f;
      s_norm[m][cls] = summed * inv;
    }
  }
  __syncthreads();

  // ---- Phase 3: logsumexp over 64 classes, combine with pos_logit ----
  if (t < 16) {
    float mx = -3.4e38f;
    for (int c2 = 0; c2 < NC; ++c2) mx = fmaxf(mx, s_norm[t][c2]);
    float s = 0.0f;
    for (int c2 = 0; c2 < NC; ++c2) s += expf(s_norm[t][c2] - mx);
    float nl = mx + logf(s);
    s_red[t] = nl - s_plog[t];
  }
  __syncthreads();
  if (t == 0) {
    float s = 0.0f;
    for (int i = 0; i < 16; ++i) s += s_red[i];
    partial[blockIdx.x] = s;
  }
}

// ---------------------------------------------------------------------------
// Kernel C: reduce 256 block partials, divide by NQ.
// ---------------------------------------------------------------------------
__global__ void __launch_bounds__(256) finalize_kernel(
    const float* __restrict__ partial, float* __restrict__ out) {
  __shared__ float red[256];
  const int t = threadIdx.x;
  red[t] = partial[t];
  __syncthreads();
  for (int s = 128; s > 0; s >>= 1) {
    if (t < s) red[t] += red[t + s];
    __syncthreads();
  }
  if (t == 0) out[0] = red[0] / (float)NQ;
}

// ---------------------------------------------------------------------------
extern "C" void kernel_launch(void* const* d_in, const int* in_sizes, int n_in,
                              void* d_out, int out_size, void* d_ws, size_t ws_size,
                              hipStream_t stream) {
  (void)in_sizes; (void)n_in; (void)out_size; (void)ws_size;
  const float* xq  = (const float*)d_in[0];
  const int*   yq  = (const int*)  d_in[1];
  const float* xs  = (const float*)d_in[2];
  const int*   ys  = (const int*)  d_in[3];
  const int*   pos = (const int*)  d_in[4];
  float* out = (float*)d_out;

  // workspace layout (floats)
  float* ws      = (float*)d_ws;
  float* mus     = ws;                   // 64*128 = 8192
  float* SS      = mus + (size_t)NC * DIM;   // 64
  float* cnts    = SS + NC;              // 64
  float* partial = cnts + NC;            // 256

  class_stats_kernel<<<NC, 128, 0, stream>>>(xs, ys, mus, SS, cnts);
  pmxe_main_kernel<<<NQ / 16, 128, 0, stream>>>(xq, yq, xs, pos, mus, SS, cnts, partial);
  finalize_kernel<<<1, 256, 0, stream>>>(partial, out);
}